// PCEN_54528904790427
// MI455X (gfx1250) — compile-verified
//
#include <hip/hip_runtime.h>
#include <math.h>

#define FLOORV 1e-12f

typedef __attribute__((ext_vector_type(2))) float v2f;
typedef __attribute__((ext_vector_type(8))) float v8f;

__device__ __forceinline__ float clamp01(float v) { return fminf(fmaxf(v, 0.f), 1.f); }

// Fast positive-base pow via hardware v_log_f32 / v_exp_f32 (base > 0 required;
// base==0 gives exp2(-inf)=0 for e>0, which matches the limit).
__device__ __forceinline__ float fpow(float base, float e) {
    return __builtin_amdgcn_exp2f(e * __builtin_amdgcn_logf(base));
}

// ---------------------------------------------------------------------------
// Generic fallback: one thread per (b,c) sequence, fully serial scan.
// ---------------------------------------------------------------------------
__global__ void pcen_serial(const float* __restrict__ x, const float* __restrict__ wts,
                            const float* __restrict__ al, const float* __restrict__ de,
                            const float* __restrict__ ro, float* __restrict__ out,
                            int Bc, int T, int C) {
    int tid = blockIdx.x * blockDim.x + threadIdx.x;
    if (tid >= Bc * C) return;
    int c = tid % C, b = tid / C;
    float w = clamp01(wts[c]);
    float a = fminf(al[c], 1.f);
    float r = fmaxf(ro[c], 1.f);
    float delta = de[c];
    float invr = 1.f / r;
    float na = -a;
    float dpow = fpow(delta, invr);
    float q = 1.f - w;
    const float* xp = x + (long long)b * T * C + c;
    float* op = out + (long long)b * T * C + c;
    float acc = xp[0];  // ema_{-1} := x0  ->  ema_0 = w*x0 + q*x0 = x0
    for (int t = 0; t < T; ++t) {
        float xv = *xp;
        acc = fmaf(w, xv, q * acc);
        float deninv = fpow(FLOORV + acc, na);   // (eps+ema)^-a, no divide
        *op = fpow(fmaf(xv, deninv, delta), invr) - dpow;
        xp += C;
        op += C;
    }
}

// ---------------------------------------------------------------------------
// Pass 1: per-(b,c,chunk) EMA partial with zero carry-in.
// ---------------------------------------------------------------------------
__global__ void pcen_partials(const float* __restrict__ x, const float* __restrict__ wts,
                              float* __restrict__ P, int Bc, int T, int C, int CL, int NCH) {
    int tid = blockIdx.x * blockDim.x + threadIdx.x;
    if (tid >= Bc * NCH * C) return;
    int c = tid % C;
    int rest = tid / C;
    int chunk = rest % NCH;
    int b = rest / NCH;
    float w = clamp01(wts[c]);
    float q = 1.f - w;
    const float* xp = x + ((long long)b * T + (long long)chunk * CL) * C + c;
    float p = 0.f;
    for (int t = 0; t < CL; ++t) {
        p = fmaf(w, *xp, q * p);
        xp += C;
    }
    P[((long long)b * C + c) * NCH + chunk] = p;
}

// ---------------------------------------------------------------------------
// Pass 2: per-sequence scan over chunk partials -> carry entering each chunk.
// ---------------------------------------------------------------------------
__global__ void pcen_scan(const float* __restrict__ x, const float* __restrict__ wts,
                          const float* __restrict__ P, float* __restrict__ carry,
                          int Bc, int T, int C, int CL, int NCH) {
    int tid = blockIdx.x * blockDim.x + threadIdx.x;
    if (tid >= Bc * C) return;
    int c = tid % C, b = tid / C;
    float w = clamp01(wts[c]);
    float q = 1.f - w;
    float A = fpow(q, (float)CL);             // q^CL (q==0 -> 0, correct)
    float acc = x[(long long)b * T * C + c];  // ema_{-1} = x[b,0,c]
    long long base = (long long)tid * NCH;
    for (int j = 0; j < NCH; ++j) {
        carry[base + j] = acc;
        acc = fmaf(A, acc, P[base + j]);
    }
}

// ---------------------------------------------------------------------------
// Pass 3: WMMA tile scan.  One wave per (b, channel-tile, chunk).
// Within a 16(time) x 16(chan) tile:
//   z_k = w * q^{-k} * x_k          (per-channel scale)
//   S   = LowerTriOnes(16x16) @ Z   (4 chained V_WMMA_F32_16X16X4_F32, C = q*carry)
//   ema_m = q^m * (S_m + q*carry)
//   out = (x*(eps+ema)^-a + delta)^(1/r) - delta^(1/r)
// ---------------------------------------------------------------------------
__global__ __launch_bounds__(32)
void pcen_wmma(const float* __restrict__ x, const float* __restrict__ wts,
               const float* __restrict__ al, const float* __restrict__ de,
               const float* __restrict__ ro, const float* __restrict__ carryws,
               float* __restrict__ out, int T, int C, int CL, int NCH) {
    __shared__ float xl[16 * 16];

    const int lane = threadIdx.x;      // 0..31 (wave32)
    const int chunk = blockIdx.x;
    const int c0 = blockIdx.y * 16;
    const int b = blockIdx.z;
    const int n = lane & 15;           // matrix column / A-row index
    const int hi = lane >> 4;          // which half of the wave
    const int c = c0 + n;

    const float w = clamp01(wts[c]);
    const float a = fminf(al[c], 1.f);
    const float r = fmaxf(ro[c], 1.f);
    const float delta = de[c];
    const float invr = 1.f / r;
    const float na = -a;
    const float dpow = fpow(delta, invr);
    const float q = 1.f - w;
    const float qsafe = fmaxf(q, 1e-30f);
    const float qinv = 1.f / qsafe;
    const float qinv2 = qinv * qinv, qinv4 = qinv2 * qinv2;
    const float q2 = q * q, q4 = q2 * q2, q8 = q4 * q4;

    // Constant lower-triangular-ones A matrix, 4 K-slices of 16x4.
    // A layout (f32 16x4): lane L -> M = L&15 ; comp0 = K (hi?2:0), comp1 = K+1.
    v2f Atri[4];
#pragma unroll
    for (int kk = 0; kk < 4; ++kk) {
        int k0 = kk * 4 + hi * 2;
        Atri[kk].x = (k0 <= n) ? 1.f : 0.f;
        Atri[kk].y = (k0 + 1 <= n) ? 1.f : 0.f;
    }

    // Per-lane z-scale factors for the 4 B slices: zscale0[kk] = w*q^{-r0},
    // zscale1[kk] = w*q^{-(r0+1)}, r0 = kk*4 + hi*2 (hoisted out of tile loop).
    float zs0[4], zs1[4];
    {
        float p = hi ? qinv2 : 1.f;
#pragma unroll
        for (int kk = 0; kk < 4; ++kk) {
            zs0[kk] = w * p;
            zs1[kk] = w * p * qinv;
            p *= qinv4;
        }
    }

    float carry = carryws[((long long)b * C + c) * NCH + chunk];

    // Incrementally advanced pointers (avoid per-store 64-bit multiplies).
    const long long tileStride = (long long)16 * C;
    const float* xrow = x + ((long long)b * T + (long long)chunk * CL) * C + c0 + n;
    float* orow = out + ((long long)b * T + (long long)chunk * CL) * C +
                  (long long)hi * 8 * C + c0 + n;

    const int ntile = CL / 16;
    for (int tt = 0; tt < ntile; ++tt) {
        // Stage the 16x16 x-tile in LDS (each half-wave loads a contiguous row).
#pragma unroll
        for (int i = 0; i < 8; ++i) {
            int row = i * 2 + hi;
            xl[row * 16 + n] = xrow[(long long)row * C];
        }
        // Prefetch next tile (global_prefetch_b8): 32 lanes cover the 16 rows.
        if (tt + 1 < ntile) {
            __builtin_prefetch(xrow - n + tileStride + (long long)(lane >> 1) * C +
                                   (lane & 1) * 8, 0, 1);
        }
        __syncthreads();

        // Accumulator pre-loaded with q*carry so D = S + q*carry.
        float qc = q * carry;
        v8f acc;
#pragma unroll
        for (int i = 0; i < 8; ++i) acc[i] = qc;

        // B layout (f32 4x16): lane L -> N = L&15 ; comp0 = row (hi?2:0), comp1 = row+1.
        // Read all 8 B values first so the DS latency overlaps the scaling VALU.
        v2f bm[4];
#pragma unroll
        for (int kk = 0; kk < 4; ++kk) {
            int r0 = kk * 4 + hi * 2;
            bm[kk].x = xl[r0 * 16 + n];
            bm[kk].y = xl[(r0 + 1) * 16 + n];
        }
#pragma unroll
        for (int kk = 0; kk < 4; ++kk) {
            bm[kk].x *= zs0[kk];
            bm[kk].y *= zs1[kk];
        }
#pragma unroll
        for (int kk = 0; kk < 4; ++kk) {
            acc = __builtin_amdgcn_wmma_f32_16x16x4_f32(
                false, Atri[kk], false, bm[kk], (short)0, acc, false, false);
        }

        // D layout: comp i -> M = hi*8 + i, N = lane&15.  ema_m = q^m * acc[i].
        float qm = hi ? q8 : 1.f;
        float emalast = 0.f;
#pragma unroll
        for (int i = 0; i < 8; ++i) {
            int mm = hi * 8 + i;
            float e = qm * acc[i];
            qm *= q;
            float xv = xl[mm * 16 + n];
            float deninv = fpow(FLOORV + e, na);   // (eps+ema)^-a, no divide
            orow[(long long)i * C] = fpow(fmaf(xv, deninv, delta), invr) - dpow;
            if (i == 7) emalast = e;
        }
        // carry for next tile = ema at M=15 (lives in the upper half-wave).
        carry = __shfl(emalast, 16 + n, 32);
        __syncthreads();

        xrow += tileStride;
        orow += tileStride;
    }
}

// ---------------------------------------------------------------------------
extern "C" void kernel_launch(void* const* d_in, const int* in_sizes, int n_in,
                              void* d_out, int out_size, void* d_ws, size_t ws_size,
                              hipStream_t stream) {
    const float* x = (const float*)d_in[0];
    const float* wts = (const float*)d_in[1];
    const float* al = (const float*)d_in[2];
    const float* de = (const float*)d_in[3];
    const float* ro = (const float*)d_in[4];
    float* out = (float*)d_out;

    const int C = in_sizes[1];
    const long long total = (long long)in_sizes[0];
    const int T = 8000;
    const int Bc = (int)(total / ((long long)T * C));

    const int CL = 160;            // chunk length (10 WMMA tiles)
    const int NCH = T / CL;        // 50 chunks

    size_t need = (size_t)2 * (size_t)Bc * C * NCH * sizeof(float);
    bool fast = (T % CL == 0) && (C % 16 == 0) && ((long long)Bc * T * C == total) &&
                (ws_size >= need) && (Bc <= 65535);

    if (!fast) {
        int nseq = Bc * C;
        pcen_serial<<<(nseq + 255) / 256, 256, 0, stream>>>(x, wts, al, de, ro, out, Bc, T, C);
        return;
    }

    float* P = (float*)d_ws;
    float* carry = P + (size_t)Bc * C * NCH;

    int n1 = Bc * NCH * C;
    pcen_partials<<<(n1 + 255) / 256, 256, 0, stream>>>(x, wts, P, Bc, T, C, CL, NCH);

    int nseq = Bc * C;
    pcen_scan<<<(nseq + 255) / 256, 256, 0, stream>>>(x, wts, P, carry, Bc, T, C, CL, NCH);

    dim3 grid(NCH, C / 16, Bc);
    pcen_wmma<<<grid, 32, 0, stream>>>(x, wts, al, de, ro, carry, out, T, C, CL, NCH);
}